// LieHomographyLayer_22067541967496
// MI455X (gfx1250) — compile-verified
//
#include <hip/hip_runtime.h>
#include <hip/hip_bf16.h>

#define NB 256

// ---------------------------------------------------------------------------
// CDNA5 async global<->LDS path (gfx1250). Guarded so the kernel still
// compiles (with a plain-VMEM fallback) if the toolchain lacks the builtins.
// Probe round 1 showed the b128 builtins take pointers to int32x4 (v4i).
// ---------------------------------------------------------------------------
#if __has_builtin(__builtin_amdgcn_global_load_async_to_lds_b128) && \
    __has_builtin(__builtin_amdgcn_global_store_async_from_lds_b128) && \
    __has_builtin(__builtin_amdgcn_s_wait_asynccnt)
#define LIE_ASYNC 1
#else
#define LIE_ASYNC 0
#endif

#if LIE_ASYNC
typedef int v4i __attribute__((vector_size(16)));
typedef __attribute__((address_space(1))) v4i* gptr_v4i;  // global int32x4*
typedef __attribute__((address_space(3))) v4i* lptr_v4i;  // LDS    int32x4*

__device__ __forceinline__ gptr_v4i as_global_v4(const void* p) {
  // On amdgcn, flat addresses of global memory equal addrspace(1) addresses.
  return (gptr_v4i)(unsigned long long)p;
}
__device__ __forceinline__ lptr_v4i as_lds_v4(void* p) {
  // generic -> LDS addrspacecast (two-step: retype, then addrspace convert)
  return (lptr_v4i)(v4i*)p;
}
#endif

// 3x3 row-major f64 matmul: c = a * b
__device__ __forceinline__ void mul3(const double* __restrict__ a,
                                     const double* __restrict__ b,
                                     double* __restrict__ c) {
#pragma unroll
  for (int i = 0; i < 3; ++i) {
#pragma unroll
    for (int j = 0; j < 3; ++j) {
      c[3 * i + j] =
          fma(a[3 * i + 0], b[0 + j],
              fma(a[3 * i + 1], b[3 + j], a[3 * i + 2] * b[6 + j]));
    }
  }
}

__global__ __launch_bounds__(NB) void lie_expm_kernel(
    const float* __restrict__ in, float* __restrict__ out, int B) {
  const int tid = threadIdx.x;
  const long long blk0 = (long long)blockIdx.x * NB;
  long long rem = (long long)B - blk0;
  const int n = (rem >= NB) ? NB : (int)rem;  // valid elements in this block

#if LIE_ASYNC
  __shared__ float s_in[NB * 8];   // 8 KB staged inputs
  __shared__ float s_out[NB * 9];  // 9 KB staged outputs
  if (tid < n) {
    const float* g = in + (blk0 + tid) * 8;
    // Each lane stages its own 32B -> only wave-local dependency.
    __builtin_amdgcn_global_load_async_to_lds_b128(
        as_global_v4(g), as_lds_v4(&s_in[tid * 8]), 0, 0);
    __builtin_amdgcn_global_load_async_to_lds_b128(
        as_global_v4(g + 4), as_lds_v4(&s_in[tid * 8 + 4]), 0, 0);
    __builtin_amdgcn_s_wait_asynccnt(0);
  }
  const float* vsrc = &s_in[tid * 8];
#else
  const float* vsrc = in + (blk0 + tid) * 8;
#endif

  float res[9];
  if (tid < n) {
    double v[8];
#pragma unroll
    for (int k = 0; k < 8; ++k) v[k] = (double)vsrc[k];

    // A = sum_k v[k] * e_k  (exact in f64: each entry is +-v_i +- v_j)
    double A[9];
    A[0] = v[3] + v[4];
    A[1] = v[5] - v[2];
    A[2] = v[0];
    A[3] = v[2] + v[5];
    A[4] = v[3] - v[4];
    A[5] = v[1];
    A[6] = v[6];
    A[7] = v[7];
    A[8] = -2.0 * v[3];

    // 1-norm (max column abs-sum) for scaling
    double c0 = fabs(A[0]) + fabs(A[3]) + fabs(A[6]);
    double c1 = fabs(A[1]) + fabs(A[4]) + fabs(A[7]);
    double c2 = fabs(A[2]) + fabs(A[5]) + fabs(A[8]);
    double nrm = fmax(c0, fmax(c1, c2));

    const double theta13 = 5.371920351148152;
    int s = 0;
    if (nrm > theta13) {
      s = (int)ceil(log2(nrm / theta13));
      if (s < 0) s = 0;
      if (s > 60) s = 60;
    }
    const double sc = ldexp(1.0, -s);
#pragma unroll
    for (int i = 0; i < 9; ++i) A[i] *= sc;

    double A2[9], A4[9], A6[9];
    mul3(A, A, A2);
    mul3(A2, A2, A4);
    mul3(A2, A4, A6);

    // Pade-13 coefficients (Higham)
    const double b0 = 64764752532480000.0, b1 = 32382376266240000.0,
                 b2 = 7771770303897600.0, b3 = 1187353796428800.0,
                 b4 = 129060195264000.0, b5 = 10559470521600.0,
                 b6 = 670442572800.0, b7 = 33522128640.0, b8 = 1323241920.0,
                 b9 = 40840800.0, b10 = 960960.0, b11 = 16380.0, b12 = 182.0,
                 b13 = 1.0;

    double W[9], T[9], U[9], V[9];
#pragma unroll
    for (int i = 0; i < 9; ++i)
      W[i] = fma(b13, A6[i], fma(b11, A4[i], b9 * A2[i]));
    mul3(A6, W, T);
#pragma unroll
    for (int i = 0; i < 9; ++i)
      T[i] += fma(b7, A6[i], fma(b5, A4[i], b3 * A2[i]));
    T[0] += b1;
    T[4] += b1;
    T[8] += b1;
    mul3(A, T, U);

#pragma unroll
    for (int i = 0; i < 9; ++i)
      W[i] = fma(b12, A6[i], fma(b10, A4[i], b8 * A2[i]));
    mul3(A6, W, V);
#pragma unroll
    for (int i = 0; i < 9; ++i)
      V[i] += fma(b6, A6[i], fma(b4, A4[i], b2 * A2[i]));
    V[0] += b0;
    V[4] += b0;
    V[8] += b0;

    double P[9], Q[9];
#pragma unroll
    for (int i = 0; i < 9; ++i) {
      P[i] = V[i] + U[i];
      Q[i] = V[i] - U[i];
    }

    // X = Q^{-1} * P  via adjugate (3x3)
    double adj[9];
    adj[0] = Q[4] * Q[8] - Q[5] * Q[7];
    adj[1] = Q[2] * Q[7] - Q[1] * Q[8];
    adj[2] = Q[1] * Q[5] - Q[2] * Q[4];
    adj[3] = Q[5] * Q[6] - Q[3] * Q[8];
    adj[4] = Q[0] * Q[8] - Q[2] * Q[6];
    adj[5] = Q[2] * Q[3] - Q[0] * Q[5];
    adj[6] = Q[3] * Q[7] - Q[4] * Q[6];
    adj[7] = Q[1] * Q[6] - Q[0] * Q[7];
    adj[8] = Q[0] * Q[4] - Q[1] * Q[3];
    const double det = Q[0] * adj[0] + Q[1] * adj[3] + Q[2] * adj[6];
    const double inv = 1.0 / det;
#pragma unroll
    for (int i = 0; i < 9; ++i) adj[i] *= inv;

    double X[9];
    mul3(adj, P, X);

    // undo scaling: square s times
    for (int k = 0; k < s; ++k) {
      double Y[9];
      mul3(X, X, Y);
#pragma unroll
      for (int i = 0; i < 9; ++i) X[i] = Y[i];
    }

    const double r = 1.0 / X[8];  // homogeneous normalization by H[2,2]
#pragma unroll
    for (int i = 0; i < 9; ++i) res[i] = (float)(X[i] * r);
  }

#if LIE_ASYNC
  if (tid < n) {
#pragma unroll
    for (int i = 0; i < 9; ++i) s_out[tid * 9 + i] = res[i];
  }
  __syncthreads();  // output chunks cross thread/wave boundaries
  float* gout = out + blk0 * 9;
  const int nbytes = n * 36;       // contiguous output bytes for this block
  const int nchunk = nbytes >> 4;  // 16B chunks
  for (int c = tid; c < nchunk; c += NB) {
    __builtin_amdgcn_global_store_async_from_lds_b128(
        as_global_v4(gout + c * 4), as_lds_v4(&s_out[c * 4]), 0, 0);
  }
  // ragged tail (dwords) if n*36 is not a multiple of 16
  for (int w = (nchunk << 2) + tid; w < (nbytes >> 2); w += NB) {
    gout[w] = s_out[w];
  }
  // s_endpgm performs an implicit wait-idle, draining ASYNCcnt stores.
#else
  if (tid < n) {
    float* gout = out + (blk0 + tid) * 9;
#pragma unroll
    for (int i = 0; i < 9; ++i) gout[i] = res[i];
  }
#endif
}

extern "C" void kernel_launch(void* const* d_in, const int* in_sizes, int n_in,
                              void* d_out, int out_size, void* d_ws,
                              size_t ws_size, hipStream_t stream) {
  const float* lie = (const float*)d_in[0];
  float* out = (float*)d_out;
  const int B = in_sizes[0] / 8;
  const int grid = (B + NB - 1) / NB;
  hipLaunchKernelGGL(lie_expm_kernel, dim3(grid), dim3(NB), 0, stream, lie,
                     out, B);
}